// HyperCrossAttention_11029476016525
// MI455X (gfx1250) — compile-verified
//
#include <hip/hip_runtime.h>
#include <math.h>

typedef __attribute__((ext_vector_type(2))) float v2f;
typedef __attribute__((ext_vector_type(8))) float v8f;

#define LNUM   8
#define KNUM   8
#define NNUM   1048576
#define PW     2048                 // n-values per wave in gram kernel
#define WAVES  8
#define CHUNK  (PW * WAVES)         // 16384 n per block
#define CHUNKS (NNUM / CHUNK)       // 64 chunks per layer-pair

// ---------------------------------------------------------------------------
// Zero the gram accumulation region of the workspace (harness poisons ws).
// ---------------------------------------------------------------------------
__global__ __launch_bounds__(512) void zero_ws_kernel(float* __restrict__ ws) {
    ws[threadIdx.x] = 0.0f;   // 512 floats: gram[L][8][8]
}

// ---------------------------------------------------------------------------
// Stage 1: Gram matrices via V_WMMA_F32_16X16X4_F32.
// Two layers (16 delta-rows) are packed into the 16 WMMA rows.
// A-fragment layout (32-bit A 16x4): lane (khalf*16 + m) holds
//   VGPR0 = A[m][2*khalf], VGPR1 = A[m][2*khalf+1]  -> one float2 load.
// For a symmetric Gram the mirrored B layout consumes the identical
// registers, so the same float2 feeds both A and B.
// Valid D elements: lanes 0-7 -> layerA Gram[r][lane] (vgpr r),
//                   lanes 24-31 -> layerB Gram[r][lane-24].
// ---------------------------------------------------------------------------
__global__ __launch_bounds__(256) void gram_kernel(const float* __restrict__ deltas,
                                                   float* __restrict__ gram) {
    const int lp    = blockIdx.x / CHUNKS;     // layer pair: 0..3
    const int chunk = blockIdx.x % CHUNKS;
    const int lane  = threadIdx.x & 31;
    const int wave  = threadIdx.x >> 5;
    const int row   = lane & 15;               // packed row: 2 layers x 8 keys
    const int khalf = lane >> 4;               // which half of the K=4 slice

    // flat row index: (2*lp + row/8)*8 + row%8 == 16*lp + row
    const float* rowptr = deltas + (size_t)(16 * lp + row) * NNUM;
    const size_t n0 = (size_t)chunk * CHUNK + (size_t)wave * PW + (size_t)(khalf * 2);

    v8f acc = {0.f, 0.f, 0.f, 0.f, 0.f, 0.f, 0.f, 0.f};

    for (int i = 0; i < PW; i += 128) {
        // gfx1250 speculative prefetch of the next strip (no counter traffic)
        __builtin_prefetch(rowptr + n0 + i + 1024, 0, 3);
        #pragma unroll
        for (int ii = 0; ii < 128; ii += 4) {
            v2f a = *(const v2f*)(rowptr + n0 + i + ii);
            // D(16x16,f32) += A(16x4,f32) x B(4x16,f32); A frag == B frag (Gram)
            acc = __builtin_amdgcn_wmma_f32_16x16x4_f32(
                      /*neg_a=*/false, a, /*neg_b=*/false, a,
                      /*c_mod=*/(short)0, acc,
                      /*reuse_a=*/false, /*reuse_b=*/false);
        }
    }

    const float scale = rsqrtf((float)NNUM);   // N^-0.5 (exactly 1/1024)

    if (lane < 8) {
        // layer 2*lp: D[m=r][j=lane], r = vgpr index
        float* g = gram + (size_t)(2 * lp) * 64 + lane;
        #pragma unroll
        for (int r = 0; r < 8; ++r)
            atomicAdd(g + r * 8, acc[r] * scale);
    } else if (lane >= 24) {
        // layer 2*lp+1: D[m=8+r][j=8+(lane-24)]
        float* g = gram + (size_t)(2 * lp + 1) * 64 + (lane - 24);
        #pragma unroll
        for (int r = 0; r < 8; ++r)
            atomicAdd(g + r * 8, acc[r] * scale);
    }
}

// ---------------------------------------------------------------------------
// Stage 2: per-(l,k) softmax over 8 logits, fold clip(beta) and the identity
// (for the "deltas +" term):  coef[l,k,j] = b*softmax(row)[j] + (j==k).
// One wave-pair, 64 threads.
// ---------------------------------------------------------------------------
__global__ __launch_bounds__(64) void softmax_coef_kernel(const float* __restrict__ gram,
                                                          const float* __restrict__ beta,
                                                          float* __restrict__ coef) {
    const int t = threadIdx.x;          // 0..63 -> (l,k)
    const int l = t >> 3;
    const int k = t & 7;

    float r[8];
    float m = -INFINITY;
    #pragma unroll
    for (int j = 0; j < 8; ++j) {
        r[j] = gram[l * 64 + k * 8 + j];
        m = fmaxf(m, r[j]);
    }
    float s = 0.0f;
    #pragma unroll
    for (int j = 0; j < 8; ++j) {
        r[j] = __expf(r[j] - m);
        s += r[j];
    }
    const float b   = fminf(fmaxf(beta[t], 0.0f), 1.0f);
    const float inv = b / s;
    #pragma unroll
    for (int j = 0; j < 8; ++j)
        coef[l * 64 + k * 8 + j] = r[j] * inv + ((j == k) ? 1.0f : 0.0f);
}

// ---------------------------------------------------------------------------
// Stage 3: streaming apply (memory-bound, ~2 FLOP/byte):
//   out[l,k,n] = last[l,k,n] + sum_j coef[l,k,j] * deltas[l,j,n]
// Each thread owns 4 consecutive n for one layer; 8x float4 delta loads are
// reused across all 8 k outputs. Coefs broadcast via LDS.
// ---------------------------------------------------------------------------
__global__ __launch_bounds__(256) void apply_kernel(const float* __restrict__ lastp,
                                                    const float* __restrict__ deltas,
                                                    const float* __restrict__ coef,
                                                    float* __restrict__ out) {
    __shared__ float c[64];
    const int l = blockIdx.y;
    const int t = threadIdx.x;
    if (t < 64) c[t] = coef[l * 64 + t];
    __syncthreads();

    const size_t n    = ((size_t)blockIdx.x * 256 + t) * 4;
    const size_t base = (size_t)l * KNUM * NNUM + n;

    float4 d[8];
    #pragma unroll
    for (int j = 0; j < 8; ++j)
        d[j] = *(const float4*)(deltas + base + (size_t)j * NNUM);

    #pragma unroll
    for (int k = 0; k < 8; ++k) {
        float4 o = *(const float4*)(lastp + base + (size_t)k * NNUM);
        #pragma unroll
        for (int j = 0; j < 8; ++j) {
            const float w = c[k * 8 + j];
            o.x = fmaf(w, d[j].x, o.x);
            o.y = fmaf(w, d[j].y, o.y);
            o.z = fmaf(w, d[j].z, o.z);
            o.w = fmaf(w, d[j].w, o.w);
        }
        *(float4*)(out + base + (size_t)k * NNUM) = o;
    }
}

// ---------------------------------------------------------------------------
extern "C" void kernel_launch(void* const* d_in, const int* in_sizes, int n_in,
                              void* d_out, int out_size, void* d_ws, size_t ws_size,
                              hipStream_t stream) {
    const float* lastp  = (const float*)d_in[0];   // last_params [L,K,N]
    const float* deltas = (const float*)d_in[1];   // deltas      [L,K,N]
    const float* beta   = (const float*)d_in[2];   // beta        [L,K]
    float* out  = (float*)d_out;

    float* gram = (float*)d_ws;        // 512 floats: [L][8][8] logits accum
    float* coef = gram + 512;          // 512 floats: [L][8][8] folded weights

    // 1) zero gram accumulators (ws is poisoned by harness)
    zero_ws_kernel<<<1, 512, 0, stream>>>(gram);

    // 2) Gram via f32 WMMA: 4 layer-pairs x 64 chunks = 256 blocks, 8 waves each
    gram_kernel<<<dim3((LNUM / 2) * CHUNKS), 256, 0, stream>>>(deltas, gram);

    // 3) softmax + beta fold (tiny)
    softmax_coef_kernel<<<1, 64, 0, stream>>>(gram, beta, coef);

    // 4) streaming apply: 1024 x 8 blocks, 256 threads, float4 per thread
    dim3 g3(NNUM / (256 * 4), LNUM);
    apply_kernel<<<g3, 256, 0, stream>>>(lastp, deltas, coef, out);
}